// Sampling_76192719831231
// MI455X (gfx1250) — compile-verified
//
#include <hip/hip_runtime.h>
#include <stdint.h>

#define NPTS  65536
#define BATCH 32
#define NSAMP 512
#define FDIM  64
#define TPB   1024

typedef unsigned int v4u __attribute__((ext_vector_type(4)));
typedef int          v8i __attribute__((ext_vector_type(8)));
typedef int          v4i __attribute__((ext_vector_type(4)));

// ---------------------------------------------------------------------------
// Kernel 1: iterative FPS. One workgroup per batch; dist[] lives in LDS
// (256 KB of the 320 KB WGP LDS). 1024 threads = 32 wave32s; each thread owns
// 64 points processed as 16 groups of 4 (float4 dist in LDS, 3x b128 xyz).
// Two barriers per iteration; centroid is a uniform cached load by all lanes.
// ---------------------------------------------------------------------------
__global__ void __launch_bounds__(TPB) Sampling_fps_kernel(
    const float* __restrict__ xyz, int* __restrict__ idx_out)
{
  extern __shared__ unsigned char smem_raw[];
  float* dist = (float*)smem_raw;                       // NPTS floats
  float* redV = (float*)(smem_raw + NPTS * 4);          // 32 wave maxima
  int*   redI = (int*)  (smem_raw + NPTS * 4 + 32 * 4); // 32 wave argmax
  int*   curS = (int*)  (smem_raw + NPTS * 4 + 64 * 4); // block winner

  const int b = blockIdx.x;
  const int t = threadIdx.x;
  const float* __restrict__ p = xyz + (size_t)b * NPTS * 3;

  for (int i = t; i < NPTS; i += TPB) dist[i] = 1e10f;
  int cur = 0;
  __syncthreads();

  for (int s = 0; s < NSAMP; ++s) {
    if (t == 0) idx_out[b * NSAMP + s] = cur;  // scan outputs farthest BEFORE update

    // uniform-address centroid load (L0/L2 cached, overlaps per wave)
    const float* c3 = p + (size_t)cur * 3;
    const float cx = c3[0], cy = c3[1], cz = c3[2];

    float bestV = -1.0f; int bestI = 0;
#pragma unroll
    for (int k = 0; k < 16; ++k) {
      const int i0 = (t << 2) + (k << 12);     // 4 contiguous points per thread
      const float4* q = (const float4*)(p + (size_t)i0 * 3);  // 48B contiguous/lane
      float4 q0 = q[0], q1 = q[1], q2 = q[2];
      float4 dv = *(float4*)(dist + i0);       // ds_load_b128, conflict-free

      float dx, dy, dz;
      dx = q0.x - cx; dy = q0.y - cy; dz = q0.z - cz;
      float d0 = dx*dx + dy*dy + dz*dz;
      dx = q0.w - cx; dy = q1.x - cy; dz = q1.y - cz;
      float d1 = dx*dx + dy*dy + dz*dz;
      dx = q1.z - cx; dy = q1.w - cy; dz = q2.x - cz;
      float d2 = dx*dx + dy*dy + dz*dz;
      dx = q2.y - cx; dy = q2.z - cy; dz = q2.w - cz;
      float d3 = dx*dx + dy*dy + dz*dz;

      dv.x = fminf(dv.x, d0);
      dv.y = fminf(dv.y, d1);
      dv.z = fminf(dv.z, d2);
      dv.w = fminf(dv.w, d3);
      *(float4*)(dist + i0) = dv;              // ds_store_b128

      // indices ascend within a thread -> strict '>' keeps first max
      if (dv.x > bestV) { bestV = dv.x; bestI = i0;     }
      if (dv.y > bestV) { bestV = dv.y; bestI = i0 + 1; }
      if (dv.z > bestV) { bestV = dv.z; bestI = i0 + 2; }
      if (dv.w > bestV) { bestV = dv.w; bestI = i0 + 3; }
    }

    // wave32 reduction (tie-break: smaller index, matching jnp.argmax)
    for (int off = 16; off > 0; off >>= 1) {
      float ov = __shfl_down(bestV, off, 32);
      int   oi = __shfl_down(bestI, off, 32);
      if (ov > bestV || (ov == bestV && oi < bestI)) { bestV = ov; bestI = oi; }
    }
    const int wave = t >> 5;
    if ((t & 31) == 0) { redV[wave] = bestV; redI[wave] = bestI; }
    __syncthreads();                            // barrier A
    if (wave == 0) {
      bestV = redV[t & 31]; bestI = redI[t & 31];
      for (int off = 16; off > 0; off >>= 1) {
        float ov = __shfl_down(bestV, off, 32);
        int   oi = __shfl_down(bestI, off, 32);
        if (ov > bestV || (ov == bestV && oi < bestI)) { bestV = ov; bestI = oi; }
      }
      if (t == 0) curS[0] = bestI;
    }
    __syncthreads();                            // barrier B
    cur = curS[0];
  }
}

// ---------------------------------------------------------------------------
// Kernel 2: gather. One block per (batch, chunk of 16 samples).
// f-gather uses the Tensor Data Mover in GATHER mode: 16x 16-bit row indices,
// tile 16 x 64 f32 -> LDS, waited with s_wait_tensorcnt, then streamed out.
// ---------------------------------------------------------------------------
__global__ void __launch_bounds__(128) Sampling_gather_kernel(
    const float* __restrict__ xyz, const float* __restrict__ f,
    const int* __restrict__ idx,
    float* __restrict__ out_xyz, float* __restrict__ out_f)
{
  __shared__ float tile[16 * FDIM];            // 4 KB
  const int b  = blockIdx.x >> 5;
  const int s0 = (blockIdx.x & 31) * 16;
  const int* __restrict__ ip = idx + b * NSAMP + s0;
  const float* __restrict__ fb = f + (size_t)b * NPTS * FDIM;

#if __has_builtin(__builtin_amdgcn_tensor_load_to_lds) && __has_builtin(__builtin_amdgcn_s_wait_tensorcnt)
  if (threadIdx.x < 32) {                       // wave 0 only, EXEC all ones
    // pack 16 row indices as 16-bit pairs (uniform scalar loads)
    unsigned pk[8];
#pragma unroll
    for (int j = 0; j < 8; ++j) {
      unsigned lo = (unsigned)ip[2 * j]     & 0xFFFFu;
      unsigned hi = (unsigned)ip[2 * j + 1] & 0xFFFFu;
      pk[j] = lo | (hi << 16);
    }
    unsigned long long ga   = (unsigned long long)(uintptr_t)fb;
    unsigned           ldsa = (unsigned)(uintptr_t)&tile[0];

    // D# group 0: count=1, gather_mode=1, 16-bit indices, lds/global addr, type=2
    v4u g0 = { 0x80000001u,
               ldsa,
               (unsigned)ga,
               (unsigned)((ga >> 32) & 0x01FFFFFFu) | (2u << 30) };
    // D# group 1: data_size=4B, tensor_dim0=64, tensor_dim1=65536,
    //             tile_dim0=64, tile_dim1=16 valid indices, dim0_stride=64
    v8i g1 = { 0x20000,
               (FDIM << 16),
               0,
               1 | (FDIM << 16),
               16,
               FDIM,
               0, 0 };
    // D# groups 2/3: 16x 16-bit row indices
    v4i g2 = { (int)pk[0], (int)pk[1], (int)pk[2], (int)pk[3] };
    v4i g3 = { (int)pk[4], (int)pk[5], (int)pk[6], (int)pk[7] };
#if __clang_major__ >= 23
    v8i gz = { 0, 0, 0, 0, 0, 0, 0, 0 };
    __builtin_amdgcn_tensor_load_to_lds(g0, g1, g2, g3, gz, 0);
#else
    __builtin_amdgcn_tensor_load_to_lds(g0, g1, g2, g3, 0);
#endif
    __builtin_amdgcn_s_wait_tensorcnt(0);
  }
  __syncthreads();
  for (int e = threadIdx.x; e < 16 * FDIM; e += 128)
    out_f[((size_t)b * NSAMP + s0) * FDIM + e] = tile[e];
#else
  // fallback: direct vector-load gather
  for (int e = threadIdx.x; e < 16 * FDIM; e += 128) {
    int r = e >> 6, c = e & 63;
    out_f[((size_t)b * NSAMP + s0) * FDIM + e] = fb[(size_t)ip[r] * FDIM + c];
  }
#endif

  if (threadIdx.x < 48) {                       // 16 rows x 3 coords
    int r = threadIdx.x / 3, c = threadIdx.x % 3;
    out_xyz[((size_t)b * NSAMP + s0 + r) * 3 + c] =
        xyz[((size_t)b * NPTS + ip[r]) * 3 + c];
  }
}

// ---------------------------------------------------------------------------
extern "C" void kernel_launch(void* const* d_in, const int* in_sizes, int n_in,
                              void* d_out, int out_size, void* d_ws, size_t ws_size,
                              hipStream_t stream) {
  const float* xyz = (const float*)d_in[0];   // [32, 65536, 3]
  const float* f   = (const float*)d_in[1];   // [32, 65536, 64]
  float* out_xyz = (float*)d_out;                                  // [32,512,3]
  float* out_f   = (float*)d_out + (size_t)BATCH * NSAMP * 3;      // [32,512,64]
  int*   idx     = (int*)d_ws;                                     // [32,512]

  const int smem_bytes = NPTS * 4 + 1024;     // dist + reduction scratch
  hipFuncSetAttribute(reinterpret_cast<const void*>(Sampling_fps_kernel),
                      hipFuncAttributeMaxDynamicSharedMemorySize, smem_bytes);

  Sampling_fps_kernel<<<BATCH, TPB, smem_bytes, stream>>>(xyz, idx);
  Sampling_gather_kernel<<<BATCH * (NSAMP / 16), 128, 0, stream>>>(
      xyz, f, idx, out_xyz, out_f);
}